// Special_RN_51874615001751
// MI455X (gfx1250) — compile-verified
//
#include <hip/hip_runtime.h>
#include <hip/hip_bf16.h>

// Problem constants (match reference)
#define BB   32      // batch
#define CC   24      // channels
#define SS   7
#define NOBJ 49      // S*S objects
#define DD   26      // C+2
#define QQ   2472
#define RN   256
#define ROWS 2401    // NOBJ*NOBJ pair-rows per batch
#define MTILES 3     // 16-row M tiles per workgroup
#define MROWS  (16 * MTILES)               // 48 rows per workgroup
#define NT2  ((ROWS + MROWS - 1) / MROWS)  // 51 row tiles per batch
#define H1S  260     // padded LDS row stride (260 mod 64 = 4 -> conflict-free A reads)

typedef float v2f __attribute__((ext_vector_type(2)));
typedef float v8f __attribute__((ext_vector_type(8)));

// ---------------------------------------------------------------------------
// Kernel 1: per-object features (x channels + coords) -> U (x_i part), V (x_j part)
// ---------------------------------------------------------------------------
__global__ __launch_bounds__(RN) void precompute_obj(
    const float* __restrict__ x, const float* __restrict__ W1,
    float* __restrict__ U, float* __restrict__ V)
{
    const int bn = blockIdx.x;           // b*49 + n
    const int b  = bn / NOBJ;
    const int n  = bn - b * NOBJ;
    const int r  = threadIdx.x;

    __shared__ float ft[DD];
    if (r < CC)         ft[r]     = x[(b * CC + r) * NOBJ + n];
    else if (r == CC)   ft[CC]    = (float)(n / SS) * (1.0f / (SS - 1));
    else if (r == CC+1) ft[CC+1]  = (float)(n % SS) * (1.0f / (SS - 1));
    __syncthreads();

    float u = 0.f, v = 0.f;
    #pragma unroll
    for (int k = 0; k < DD; ++k) {
        const float f = ft[k];
        u += f * W1[k * RN + r];
        v += f * W1[(DD + k) * RN + r];
    }
    U[bn * RN + r] = u;
    V[bn * RN + r] = v;
}

// ---------------------------------------------------------------------------
// Kernel 2: qv[b,r] = b1[r] + sum_k qst[b,k] * W1[52+k, r]
// ---------------------------------------------------------------------------
__global__ __launch_bounds__(RN) void qv_kernel(
    const float* __restrict__ qst, const float* __restrict__ W1,
    const float* __restrict__ b1, float* __restrict__ qv)
{
    const int b = blockIdx.x;
    const int r = threadIdx.x;
    __shared__ float qs[RN];

    float acc = b1[r];
    for (int k0 = 0; k0 < QQ; k0 += RN) {
        const int kk = k0 + r;
        qs[r] = (kk < QQ) ? qst[b * QQ + kk] : 0.f;
        __syncthreads();
        const int kmax = (QQ - k0 < RN) ? (QQ - k0) : RN;
        for (int k = 0; k < kmax; ++k)
            acc += qs[k] * W1[(2 * DD + k0 + k) * RN + r];
        __syncthreads();
    }
    qv[b * RN + r] = acc;
}

// ---------------------------------------------------------------------------
// Kernel 2b: repack Wg into K-pair-major so a B fragment is one b64 load:
//   WgP[kp*256 + n] = { Wg[2kp*256 + n], Wg[(2kp+1)*256 + n] }
// ---------------------------------------------------------------------------
__global__ __launch_bounds__(RN) void repack_wg(
    const float* __restrict__ Wg, v2f* __restrict__ WgP)
{
    const int kp = blockIdx.x;           // 0..127
    const int n  = threadIdx.x;          // 0..255
    v2f p;
    p.x = Wg[(2 * kp) * RN + n];
    p.y = Wg[(2 * kp + 1) * RN + n];
    WgP[kp * RN + n] = p;
}

// ---------------------------------------------------------------------------
// Kernel 3: fused g-MLP over a 48-row pair tile (single LDS buffer, reused).
//   phase 1: buf = h1 = relu(U[q] + V[p] + qv)
//   phase 2: layer-1 accumulators for all (2 N-tiles x 3 M-tiles) in regs
//   phase 3: buf = h2 = relu(h1 @ Wg + bg)      (after barrier, overwrite)
//   phase 4: h3 = relu(h2 @ Wg + bg), masked row-sum -> atomic into xg
// Each B fragment (one b64) feeds 3 WMMAs (one per M tile).
// ---------------------------------------------------------------------------
__global__ __launch_bounds__(RN) void rn_main(
    const float* __restrict__ U, const float* __restrict__ V,
    const float* __restrict__ qv, const v2f* __restrict__ WgP,
    const float* __restrict__ bg, float* __restrict__ xg)
{
    const int t   = blockIdx.x;     // row tile within batch
    const int b   = blockIdx.y;
    const int m0  = t * MROWS;
    const int tid = threadIdx.x;

    __shared__ float buf[MROWS * H1S];   // 48*260*4 = 49920 B

    // ---- phase 1: build h1 tile; thread owns column r = tid ----
    {
        const float qvr = qv[b * RN + tid];
        for (int m = 0; m < MROWS; ++m) {
            const int row = m0 + m;
            float val = 0.f;
            if (row < ROWS) {
                const int p = row / NOBJ;
                const int q = row - p * NOBJ;
                val = U[(b * NOBJ + q) * RN + tid]
                    + V[(b * NOBJ + p) * RN + tid] + qvr;
                val = fmaxf(val, 0.f);
            }
            buf[m * H1S + tid] = val;
        }
    }
    __syncthreads();

    const int wave = tid >> 5;
    const int lane = tid & 31;
    const int ml   = lane & 15;   // A-row / B-col / D-col sub-index
    const int kh   = lane >> 4;   // K-pair select (A/B) / M+8 half (C/D)

    // ---- phase 2: layer g1, all accumulators in registers ----
    v8f acc[2][MTILES];
    #pragma unroll
    for (int nt = 0; nt < 2; ++nt) {
        const int nn = wave * 32 + nt * 16 + ml;
        v8f c0 = {}, c1 = {}, c2 = {};
        for (int k0 = 0; k0 < RN; k0 += 4) {
            const int k  = k0 + 2 * kh;
            const v2f bb = WgP[((k0 >> 1) + kh) * RN + nn];      // B: K=k,k+1 N=nn
            const v2f a0 = *(const v2f*)&buf[(ml     ) * H1S + k];
            const v2f a1 = *(const v2f*)&buf[(ml + 16) * H1S + k];
            const v2f a2 = *(const v2f*)&buf[(ml + 32) * H1S + k];
            c0 = __builtin_amdgcn_wmma_f32_16x16x4_f32(false, a0, false, bb, (short)0, c0, false, false);
            c1 = __builtin_amdgcn_wmma_f32_16x16x4_f32(false, a1, false, bb, (short)0, c1, false, false);
            c2 = __builtin_amdgcn_wmma_f32_16x16x4_f32(false, a2, false, bb, (short)0, c2, false, false);
        }
        acc[nt][0] = c0; acc[nt][1] = c1; acc[nt][2] = c2;
    }
    __syncthreads();   // all h1 reads complete -> safe to overwrite buf

    // ---- phase 3: write h2 = relu(acc + bg) into buf ----
    #pragma unroll
    for (int nt = 0; nt < 2; ++nt) {
        const int nn  = wave * 32 + nt * 16 + ml;
        const float bgn = bg[nn];
        #pragma unroll
        for (int mt = 0; mt < MTILES; ++mt) {
            #pragma unroll
            for (int j = 0; j < 8; ++j)             // D: M = mt*16 + j + 8*kh
                buf[(mt * 16 + j + 8 * kh) * H1S + nn] =
                    fmaxf(acc[nt][mt][j] + bgn, 0.f);
        }
    }
    __syncthreads();

    // ---- phase 4: layer g2 + masked row reduction into xg ----
    #pragma unroll
    for (int nt = 0; nt < 2; ++nt) {
        const int nn = wave * 32 + nt * 16 + ml;
        v8f c0 = {}, c1 = {}, c2 = {};
        for (int k0 = 0; k0 < RN; k0 += 4) {
            const int k  = k0 + 2 * kh;
            const v2f bb = WgP[((k0 >> 1) + kh) * RN + nn];
            const v2f a0 = *(const v2f*)&buf[(ml     ) * H1S + k];
            const v2f a1 = *(const v2f*)&buf[(ml + 16) * H1S + k];
            const v2f a2 = *(const v2f*)&buf[(ml + 32) * H1S + k];
            c0 = __builtin_amdgcn_wmma_f32_16x16x4_f32(false, a0, false, bb, (short)0, c0, false, false);
            c1 = __builtin_amdgcn_wmma_f32_16x16x4_f32(false, a1, false, bb, (short)0, c1, false, false);
            c2 = __builtin_amdgcn_wmma_f32_16x16x4_f32(false, a2, false, bb, (short)0, c2, false, false);
        }
        const float bgn = bg[nn];
        float s = 0.f;
        #pragma unroll
        for (int j = 0; j < 8; ++j) {
            const int M = j + 8 * kh;
            if (m0 + M      < ROWS) s += fmaxf(c0[j] + bgn, 0.f);
            if (m0 + 16 + M < ROWS) s += fmaxf(c1[j] + bgn, 0.f);
            if (m0 + 32 + M < ROWS) s += fmaxf(c2[j] + bgn, 0.f);
        }
        s += __shfl_xor(s, 16);                  // combine lane halves (same N)
        if (lane < 16)
            atomicAdd(&xg[b * RN + nn], s);
    }
}

// ---------------------------------------------------------------------------
// Kernel 4: f MLP (2 layers, shared Wf/bf):  out = relu(relu(xg@Wf+bf)@Wf+bf)
// ---------------------------------------------------------------------------
__global__ __launch_bounds__(RN) void f_mlp(
    const float* __restrict__ xg, const float* __restrict__ Wf,
    const float* __restrict__ bf, float* __restrict__ out)
{
    const int b = blockIdx.x;
    const int r = threadIdx.x;
    __shared__ float xs[RN];
    __shared__ float y[RN];

    xs[r] = xg[b * RN + r];
    __syncthreads();

    float a1 = bf[r];
    for (int k = 0; k < RN; ++k) a1 += xs[k] * Wf[k * RN + r];
    y[r] = fmaxf(a1, 0.f);
    __syncthreads();

    float a2 = bf[r];
    for (int k = 0; k < RN; ++k) a2 += y[k] * Wf[k * RN + r];
    out[b * RN + r] = fmaxf(a2, 0.f);
}

// ---------------------------------------------------------------------------
extern "C" void kernel_launch(void* const* d_in, const int* in_sizes, int n_in,
                              void* d_out, int out_size, void* d_ws, size_t ws_size,
                              hipStream_t stream)
{
    const float* x   = (const float*)d_in[0];
    const float* qst = (const float*)d_in[1];
    const float* W1  = (const float*)d_in[2];
    const float* b1  = (const float*)d_in[3];
    const float* Wg  = (const float*)d_in[4];
    const float* bg  = (const float*)d_in[5];
    const float* Wf  = (const float*)d_in[6];
    const float* bf  = (const float*)d_in[7];
    // d_in[8], d_in[9]: g_layers=2, f_layers=2 (fixed in reference; hardcoded)
    float* out = (float*)d_out;

    // workspace layout (floats)
    float* U   = (float*)d_ws;                // B*49*256
    float* V   = U   + BB * NOBJ * RN;        // B*49*256
    float* qv  = V   + BB * NOBJ * RN;        // B*256
    float* xg  = qv  + BB * RN;               // B*256 (accumulator)
    v2f*   WgP = (v2f*)(xg + BB * RN);        // 128*256 float2 (repacked Wg)

    hipMemsetAsync(xg, 0, BB * RN * sizeof(float), stream);

    precompute_obj<<<dim3(BB * NOBJ), dim3(RN), 0, stream>>>(x, W1, U, V);
    qv_kernel<<<dim3(BB), dim3(RN), 0, stream>>>(qst, W1, b1, qv);
    repack_wg<<<dim3(RN / 2), dim3(RN), 0, stream>>>(Wg, WgP);
    rn_main<<<dim3(NT2, BB), dim3(RN), 0, stream>>>(U, V, qv, WgP, bg, xg);
    f_mlp<<<dim3(BB), dim3(RN), 0, stream>>>(xg, Wf, bf, out);
}